// KFRNN_50233937494278
// MI455X (gfx1250) — compile-verified
//
#include <hip/hip_runtime.h>

// Kalman-filter scan, B=4096 batches, N=4, M=2, S=512.
// One lane per batch; filter state kept in VGPRs; measurements double-buffered
// in LDS. Staging uses the gfx1250 Tensor Data Mover (one tensor_load_to_lds
// descriptor per chunk, with hardware LDS padding giving a conflict-free
// 9-float row stride) when the builtin exists, else async global->LDS b32
// copies. Outputs staged 4 steps and written as 16B non-temporal stores.

namespace {
constexpr int kB = 4096;
constexpr int kN = 4;
constexpr int kM = 2;
constexpr int kS = 512;

constexpr int BLK  = 128;           // threads per block == batches per block
constexpr int TCH  = 8;             // steps staged per chunk
constexpr int NCH  = kS / TCH;      // 64 chunks
constexpr int ROWS = BLK * kM;      // 256 measurement rows per block
constexpr int LROW = TCH + 1;       // 9 floats: odd stride -> bank-conflict-free
constexpr int BUFW = ROWS * LROW;   // words per LDS buffer
constexpr int LOADS_PER_CHUNK = ROWS * TCH / BLK;  // 16 async b32 per thread
}

typedef float v4f __attribute__((ext_vector_type(4)));

#define KF_HAS_TDM __has_builtin(__builtin_amdgcn_tensor_load_to_lds)
#define KF_HAS_WAIT_TENSOR __has_builtin(__builtin_amdgcn_s_wait_tensorcnt)

#if KF_HAS_TDM
typedef unsigned u32x4 __attribute__((ext_vector_type(4)));
typedef int      i32x8 __attribute__((ext_vector_type(8)));
typedef int      i32x4 __attribute__((ext_vector_type(4)));

// Issue one TDM descriptor: 2D tile (TCH floats x ROWS rows) from the
// (kB*kM x kS) f32 measurement tensor into LDS, with 1-DWORD padding after
// every 8 DWORDs (-> LROW = 9-float rows in LDS).
__device__ __forceinline__ void tdm_issue_chunk(const float* meas, unsigned row0,
                                                unsigned smbase, int c) {
  const unsigned long long gaddr =
      (unsigned long long)(size_t)meas +
      ((unsigned long long)row0 * (unsigned)kS + (unsigned)(c * TCH)) * 4ull;
  const unsigned lds = smbase + (unsigned)((c & 1) * BUFW * 4);

  // ---- D# group 0 (ISA 8.3): count=1, lds_addr, global_addr, type=2 ----
  u32x4 g0;
  g0.x = 1u;                                    // count=1, user mode, no gather
  g0.y = lds;                                   // [63:32]  lds_addr (bytes)
  g0.z = (unsigned)gaddr;                       // [95:64]  global_addr[31:0]
  g0.w = (unsigned)((gaddr >> 32) & 0x01FFFFFFull) | (2u << 30); // addr[56:32], type=2

  // ---- D# group 1 (ISA 8.4) ----
  i32x8 g1;
  g1[0] = (2 << 16)      // data_size = 4 bytes
        | (1 << 20)      // pad_enable
        | (2 << 22);     // pad_interval: 8 DWORDs   (pad_amount=0 -> 1 DWORD)
  g1[1] = (int)(((unsigned)kS & 0xFFFFu) << 16);          // tensor_dim0 lo16 in [63:48]
  g1[2] = (int)((((unsigned)(kB * kM)) & 0xFFFFu) << 16); // dim0 hi16 (=0) | tensor_dim1 lo16
  g1[3] = (int)((unsigned)TCH << 16);                     // tensor_dim1 hi16 (=0) | tile_dim0
  g1[4] = ROWS;                                           // tile_dim1 | tile_dim2(=0)
  g1[5] = kS;                                             // tensor_dim0_stride[31:0]
  g1[6] = 0;                                              // stride0 hi | stride1 lo
  g1[7] = 0;                                              // stride1 hi
  const i32x4 gz4 = {0, 0, 0, 0};            // groups 2/3: dims beyond 2D unused
  const i32x8 gz8 = {0, 0, 0, 0, 0, 0, 0, 0};
  // amdgpu-toolchain (clang-23) 6-arg form: (g0, g1, g2, g3, g4, cpol)
  __builtin_amdgcn_tensor_load_to_lds(g0, g1, gz4, gz4, gz8, 0);
}
#endif  // KF_HAS_TDM

__device__ __forceinline__ void async_ld_b32(unsigned lds_byte, unsigned g_byte,
                                             const float* base) {
  // GVS mode: mem = SGPR64 base + VGPR32 offset; VDST VGPR = LDS byte address.
  asm volatile("global_load_async_to_lds_b32 %0, %1, %2"
               :
               : "v"(lds_byte), "v"(g_byte), "s"(base)
               : "memory");
}

__device__ __forceinline__ void wait_tensor(int n) {
#if KF_HAS_TDM
#if KF_HAS_WAIT_TENSOR
  if (n == 0) __builtin_amdgcn_s_wait_tensorcnt(0);
  else        __builtin_amdgcn_s_wait_tensorcnt(1);
#else
  if (n == 0) asm volatile("s_wait_tensorcnt 0" ::: "memory");
  else        asm volatile("s_wait_tensorcnt 1" ::: "memory");
#endif
#else
  (void)n;
#endif
}

__global__ __launch_bounds__(BLK)
void kf_scan_kernel(const float* __restrict__ meas,   // (B, M, S)
                    const float* __restrict__ x0,     // (B, N)
                    const float* __restrict__ P0,     // (B, N, N)
                    const float* __restrict__ Fm,     // (N, N)
                    const float* __restrict__ Qm,     // (N, N)
                    const float* __restrict__ Hm,     // (M, N)
                    const float* __restrict__ Rm,     // (M, M)
                    float* __restrict__ out) {        // preds | states | covs
  __shared__ float sm[2 * BUFW];

  const int tid = (int)threadIdx.x;
  const int b   = (int)blockIdx.x * BLK + tid;
  const unsigned row0   = (unsigned)((int)blockIdx.x * ROWS);
  const unsigned smbase = (unsigned)(size_t)(&sm[0]);  // low 32 bits = LDS offset

  // Uniform model matrices (uniform addresses -> scalar loads).
  float F[4][4], Q[4][4], H[2][4], R[2][2];
  #pragma unroll
  for (int i = 0; i < 16; ++i) { (&F[0][0])[i] = Fm[i]; }
  #pragma unroll
  for (int i = 0; i < 16; ++i) { (&Q[0][0])[i] = Qm[i]; }
  #pragma unroll
  for (int i = 0; i < 8; ++i)  { (&H[0][0])[i] = Hm[i]; }
  #pragma unroll
  for (int i = 0; i < 4; ++i)  { (&R[0][0])[i] = Rm[i]; }

  // Per-batch filter state lives entirely in VGPRs.
  float x[4], P[4][4];
  #pragma unroll
  for (int i = 0; i < 4; ++i)  { x[i] = x0[b * 4 + i]; }
  #pragma unroll
  for (int i = 0; i < 16; ++i) { (&P[0][0])[i] = P0[b * 16 + i]; }

  float* outPred = out;                                // (B, M, S)
  float* outX    = out + (size_t)kB * kM * kS;         // (B, N, S)
  float* outC    = outX + (size_t)kB * kN * kS;        // (B, N, N, S)

#if KF_HAS_TDM
  const bool issuer = (tid < 32);  // one wave per workgroup drives the TDM
  if (issuer) { tdm_issue_chunk(meas, row0, smbase, 0); }
#else
  auto issue_chunk = [&](int c) {
    const unsigned s0   = (unsigned)(c * TCH);
    const unsigned bufb = smbase + (unsigned)((c & 1) * BUFW * 4);
    #pragma unroll
    for (int i = 0; i < LOADS_PER_CHUNK; ++i) {
      const unsigned li  = (unsigned)(i * BLK + tid);
      const unsigned row = li >> 3;        // TCH == 8: 8 lanes cover one row chunk
      const unsigned ss  = li & 7u;        // -> 32B-contiguous per 8 lanes
      const unsigned gb  = ((row0 + row) * (unsigned)kS + s0 + ss) * 4u;
      const unsigned lb  = bufb + (row * (unsigned)LROW + ss) * 4u;
      async_ld_b32(lb, gb, meas);
    }
  };
  issue_chunk(0);
#endif

  for (int c = 0; c < NCH; ++c) {
#if KF_HAS_TDM
    if (issuer) {
      if (c + 1 < NCH) {
        tdm_issue_chunk(meas, row0, smbase, c + 1);  // prefetch next chunk
        wait_tensor(1);   // tensor ops complete in order -> chunk c is resident
      } else {
        wait_tensor(0);
      }
    }
#else
    if (c + 1 < NCH) {
      issue_chunk(c + 1);
      asm volatile("s_wait_asynccnt 16" ::: "memory");
    } else {
      asm volatile("s_wait_asynccnt 0" ::: "memory");
    }
#endif
    __syncthreads();           // chunk-c data visible to all waves

    const float* z0p = &sm[(c & 1) * BUFW + (tid * 2 + 0) * LROW];
    const float* z1p = z0p + LROW;

    #pragma unroll
    for (int g = 0; g < TCH / 4; ++g) {
      float pb[2][4], xb[4][4], cb[16][4];

      #pragma unroll
      for (int q = 0; q < 4; ++q) {
        const int sl = g * 4 + q;
        const float z0 = z0p[sl];
        const float z1 = z1p[sl];

        // ---- predict: x_pred = F x ; P_pred = F P F^T + Q ----
        float xp[4];
        #pragma unroll
        for (int i = 0; i < 4; ++i) {
          xp[i] = F[i][0]*x[0] + F[i][1]*x[1] + F[i][2]*x[2] + F[i][3]*x[3];
        }
        float FP[4][4];
        #pragma unroll
        for (int i = 0; i < 4; ++i) {
          #pragma unroll
          for (int k = 0; k < 4; ++k) {
            FP[i][k] = F[i][0]*P[0][k] + F[i][1]*P[1][k] + F[i][2]*P[2][k] + F[i][3]*P[3][k];
          }
        }
        float Pp[4][4];
        #pragma unroll
        for (int i = 0; i < 4; ++i) {
          #pragma unroll
          for (int l = 0; l < 4; ++l) {
            Pp[i][l] = Q[i][l] + FP[i][0]*F[l][0] + FP[i][1]*F[l][1]
                               + FP[i][2]*F[l][2] + FP[i][3]*F[l][3];
          }
        }

        // ---- innovation ----
        const float pr0 = H[0][0]*xp[0] + H[0][1]*xp[1] + H[0][2]*xp[2] + H[0][3]*xp[3];
        const float pr1 = H[1][0]*xp[0] + H[1][1]*xp[1] + H[1][2]*xp[2] + H[1][3]*xp[3];
        const float y0 = z0 - pr0;
        const float y1 = z1 - pr1;

        float HP[2][4];
        #pragma unroll
        for (int a = 0; a < 2; ++a) {
          #pragma unroll
          for (int k = 0; k < 4; ++k) {
            HP[a][k] = H[a][0]*Pp[0][k] + H[a][1]*Pp[1][k] + H[a][2]*Pp[2][k] + H[a][3]*Pp[3][k];
          }
        }
        const float S00 = R[0][0] + HP[0][0]*H[0][0] + HP[0][1]*H[0][1] + HP[0][2]*H[0][2] + HP[0][3]*H[0][3];
        const float S01 = R[0][1] + HP[0][0]*H[1][0] + HP[0][1]*H[1][1] + HP[0][2]*H[1][2] + HP[0][3]*H[1][3];
        const float S10 = R[1][0] + HP[1][0]*H[0][0] + HP[1][1]*H[0][1] + HP[1][2]*H[0][2] + HP[1][3]*H[0][3];
        const float S11 = R[1][1] + HP[1][0]*H[1][0] + HP[1][1]*H[1][1] + HP[1][2]*H[1][2] + HP[1][3]*H[1][3];

        float PHt[4][2];
        #pragma unroll
        for (int i = 0; i < 4; ++i) {
          PHt[i][0] = Pp[i][0]*H[0][0] + Pp[i][1]*H[0][1] + Pp[i][2]*H[0][2] + Pp[i][3]*H[0][3];
          PHt[i][1] = Pp[i][0]*H[1][0] + Pp[i][1]*H[1][1] + Pp[i][2]*H[1][2] + Pp[i][3]*H[1][3];
        }

        // ---- K = PHt * S^{-1} (2x2 adjugate solve, matches solve(S, PHt^T)^T) ----
        const float det  = S00 * S11 - S01 * S10;
        const float rdet = 1.0f / det;
        float K[4][2];
        #pragma unroll
        for (int i = 0; i < 4; ++i) {
          K[i][0] = ( S11 * PHt[i][0] - S01 * PHt[i][1]) * rdet;
          K[i][1] = (-S10 * PHt[i][0] + S00 * PHt[i][1]) * rdet;
        }

        #pragma unroll
        for (int i = 0; i < 4; ++i) {
          x[i] = xp[i] + K[i][0]*y0 + K[i][1]*y1;
        }

        // ---- P = (I - K H) P_pred ----
        float A[4][4];
        #pragma unroll
        for (int i = 0; i < 4; ++i) {
          #pragma unroll
          for (int j = 0; j < 4; ++j) {
            A[i][j] = ((i == j) ? 1.0f : 0.0f) - (K[i][0]*H[0][j] + K[i][1]*H[1][j]);
          }
        }
        #pragma unroll
        for (int i = 0; i < 4; ++i) {
          #pragma unroll
          for (int l = 0; l < 4; ++l) {
            P[i][l] = A[i][0]*Pp[0][l] + A[i][1]*Pp[1][l] + A[i][2]*Pp[2][l] + A[i][3]*Pp[3][l];
          }
        }

        // ---- stage outputs for 4-step vector store ----
        pb[0][q] = pr0; pb[1][q] = pr1;
        #pragma unroll
        for (int n = 0; n < 4; ++n)  { xb[n][q] = x[n]; }
        #pragma unroll
        for (int r = 0; r < 16; ++r) { cb[r][q] = (&P[0][0])[r]; }
      }

      const int sg = c * TCH + g * 4;
      #pragma unroll
      for (int m = 0; m < 2; ++m) {
        v4f v = { pb[m][0], pb[m][1], pb[m][2], pb[m][3] };
        __builtin_nontemporal_store(v, (v4f*)(outPred + ((size_t)b * kM + m) * kS + sg));
      }
      #pragma unroll
      for (int n = 0; n < 4; ++n) {
        v4f v = { xb[n][0], xb[n][1], xb[n][2], xb[n][3] };
        __builtin_nontemporal_store(v, (v4f*)(outX + ((size_t)b * kN + n) * kS + sg));
      }
      #pragma unroll
      for (int r = 0; r < 16; ++r) {
        v4f v = { cb[r][0], cb[r][1], cb[r][2], cb[r][3] };
        __builtin_nontemporal_store(v, (v4f*)(outC + ((size_t)b * 16 + r) * kS + sg));
      }
    }
    __syncthreads();  // protect buffer (c&1) before it is refilled for chunk c+2
  }
}

extern "C" void kernel_launch(void* const* d_in, const int* in_sizes, int n_in,
                              void* d_out, int out_size, void* d_ws, size_t ws_size,
                              hipStream_t stream) {
  (void)in_sizes; (void)n_in; (void)d_ws; (void)ws_size; (void)out_size;
  const float* meas = (const float*)d_in[0];
  const float* x0   = (const float*)d_in[1];
  const float* P0   = (const float*)d_in[2];
  const float* Fm   = (const float*)d_in[3];
  const float* Qm   = (const float*)d_in[4];
  const float* Hm   = (const float*)d_in[5];
  const float* Rm   = (const float*)d_in[6];
  float* out = (float*)d_out;

  dim3 grid(kB / BLK), block(BLK);
  hipLaunchKernelGGL(kf_scan_kernel, grid, block, 0, stream,
                     meas, x0, P0, Fm, Qm, Hm, Rm, out);
}